// QuantizeEMAReset_5652176961855
// MI455X (gfx1250) — compile-verified
//
#include <hip/hip_runtime.h>
#include <hip/hip_bf16.h>
#include <math.h>

typedef __attribute__((ext_vector_type(2))) float v2f;
typedef __attribute__((ext_vector_type(4))) float f4;
typedef __attribute__((ext_vector_type(8))) float v8f;

#define NB_CODE 1024
#define CODE_DIM 64
#define NT (32 * 8192)            // 262144 rows
#define ROWS_PER_BLOCK 128
#define NBLOCKS (NT / ROWS_PER_BLOCK)   // 2048
#define NTILES (NB_CODE / 16)           // 64 code tiles of 16
#define PITCH 68                  // LDS row pitch (floats): conflict-free frag reads

// ---------------------------------------------------------------------------
// Kernel 1: per-code squared norms ||k_j||^2  (fixed order -> deterministic)
// ---------------------------------------------------------------------------
__global__ void knorm_kernel(const float* __restrict__ cb, float* __restrict__ kn) {
  int j = blockIdx.x * blockDim.x + threadIdx.x;
  if (j < NB_CODE) {
    const f4* row = (const f4*)(cb + j * CODE_DIM);
    float s = 0.f;
#pragma unroll
    for (int q = 0; q < CODE_DIM / 4; ++q) {
      f4 v = row[q];
      s += v.x * v.x + v.y * v.y + v.z * v.z + v.w * v.w;
    }
    kn[j] = s;
  }
}

// ---------------------------------------------------------------------------
// Kernel 2: WMMA distance GEMM + argmin + gather + histogram + commit partials
// block = 256 threads = 8 waves; wave handles 16 rows; block handles 128 rows
// ---------------------------------------------------------------------------
__launch_bounds__(256)
__global__ void vq_argmin_kernel(const float* __restrict__ x,
                                 const float* __restrict__ cb,
                                 const float* __restrict__ knorm,
                                 float* __restrict__ xout,
                                 unsigned* __restrict__ counts,
                                 float* __restrict__ partial) {
  __shared__ float ldsX[ROWS_PER_BLOCK * PITCH];   // 34.0 KB, padded
  __shared__ float ldsB[16 * PITCH];               //  4.3 KB, padded
  __shared__ float ldsKn[NB_CODE];                 //  4.0 KB
  __shared__ int   ldsIdx[ROWS_PER_BLOCK];
  __shared__ float ldsRed[256];

  const int t    = threadIdx.x;
  const int lane = t & 31;
  const int wave = t >> 5;
  const int lm   = lane & 15;   // A: row m ; B: col n ; C: col n
  const int hi   = lane >> 4;   // A/B: K-pair select ; C: row half select
  const int rowBase = blockIdx.x * ROWS_PER_BLOCK;

  // ---- stage X tile (coalesced float4) into padded LDS ----
  {
    const f4* gx = (const f4*)(x + (size_t)rowBase * CODE_DIM);
#pragma unroll
    for (int i = 0; i < 8; ++i) {
      int e4 = i * 256 + t;          // 0..2047 float4s
      int r  = e4 >> 4;              // 16 float4 per row
      int q4 = e4 & 15;
      f4 v = gx[e4];
      *(f4*)&ldsX[r * PITCH + q4 * 4] = v;
    }
  }
  // ---- stage code norms ----
  {
    f4 v = ((const f4*)knorm)[t];
    *(f4*)&ldsKn[t * 4] = v;
  }
  __syncthreads();

  // ---- build A fragments (16x4 fp32 layout, ISA 7.12.2), pre-scaled by -2 ----
  // lane holds row m=lm; VGPR pair c covers K = 4c + 2*hi .. +1
  v2f a[16];
  {
    const float* xrow = &ldsX[(wave * 16 + lm) * PITCH];
#pragma unroll
    for (int c = 0; c < 16; ++c) {
      v2f v = *(const v2f*)&xrow[4 * c + 2 * hi];
      a[c] = v * -2.0f;
    }
  }

  float best[8];
  int   bidx[8];
#pragma unroll
  for (int v = 0; v < 8; ++v) { best[v] = 3.402823e38f; bidx[v] = 0; }

  // ---- loop over 64 codebook tiles of 16 codes ----
  for (int nt = 0; nt < NTILES; ++nt) {
    __syncthreads();   // previous tile's consumers done before overwrite
    {                  // stage 16x64 codebook tile: one float4 per thread
      int j  = t >> 4;           // code within tile
      int q4 = t & 15;
      f4 v = *(const f4*)(cb + (size_t)(nt * 16 + j) * CODE_DIM + q4 * 4);
      *(f4*)&ldsB[j * PITCH + q4 * 4] = v;
    }
    __syncthreads();

    // D = (-2 X) * K^T  accumulated over K=64 via 16 chained fp32 WMMAs
    v8f acc = {};
    const float* brow = &ldsB[lm * PITCH];
#pragma unroll
    for (int c = 0; c < 16; ++c) {
      v2f b = *(const v2f*)&brow[4 * c + 2 * hi];
      acc = __builtin_amdgcn_wmma_f32_16x16x4_f32(
          false, a[c], false, b, (short)0, acc, false, false);
    }

    // dist = ||k||^2 - 2 x.k ; per lane: column n = nt*16+lm, rows v+8*hi
    float kn   = ldsKn[nt * 16 + lm];
    int   code = nt * 16 + lm;
#pragma unroll
    for (int v = 0; v < 8; ++v) {
      float d = kn + acc[v];
      if (d < best[v]) { best[v] = d; bidx[v] = code; }   // codes ascend -> first-min kept
    }
  }

  // ---- argmin across the 16 columns (lanes of the half-wave), per row v ----
#pragma unroll
  for (int v = 0; v < 8; ++v) {
    float bv = best[v];
    int   bi = bidx[v];
#pragma unroll
    for (int s = 1; s < 16; s <<= 1) {
      float ov = __shfl_xor(bv, s, 32);
      int   oi = __shfl_xor(bi, s, 32);
      if (ov < bv || (ov == bv && oi < bi)) { bv = ov; bi = oi; }  // jnp.argmin tie rule
    }
    if (lm == v) {                      // one writer per row
      int row = wave * 16 + v + 8 * hi;
      ldsIdx[row] = bi;
      atomicAdd(&counts[bi], 1u);       // integer -> exact & deterministic
    }
  }
  __syncthreads();

  // ---- gather x_d = codebook[idx], write output, accumulate commit partial ----
  float csum = 0.f;
#pragma unroll
  for (int i = 0; i < 8; ++i) {
    int e4 = i * 256 + t;
    int r  = e4 >> 4;
    int q4 = e4 & 15;
    int idx = ldsIdx[r];
    f4 xd = *(const f4*)(cb + (size_t)idx * CODE_DIM + q4 * 4);
    f4 xv = *(const f4*)&ldsX[r * PITCH + q4 * 4];
    *(f4*)&xout[(size_t)(rowBase + r) * CODE_DIM + q4 * 4] = xd;  // STE output == x_d
    f4 df = xv - xd;
    csum += df.x * df.x + df.y * df.y + df.z * df.z + df.w * df.w;
  }
  ldsRed[t] = csum;
  __syncthreads();
  for (int s = 128; s > 0; s >>= 1) {   // fixed tree -> deterministic
    if (t < s) ldsRed[t] += ldsRed[t + s];
    __syncthreads();
  }
  if (t == 0) partial[blockIdx.x] = ldsRed[0];
}

// ---------------------------------------------------------------------------
// Kernel 3: finalize commit loss + perplexity (single block, fixed order)
// ---------------------------------------------------------------------------
__global__ void finalize_kernel(const float* __restrict__ partial,
                                const unsigned* __restrict__ counts,
                                float* __restrict__ out) {
  __shared__ float red[256];
  int t = threadIdx.x;

  float s = 0.f;
  for (int i = t; i < NBLOCKS; i += 256) s += partial[i];
  red[t] = s;
  __syncthreads();
  for (int k = 128; k > 0; k >>= 1) { if (t < k) red[t] += red[t + k]; __syncthreads(); }
  float commit = red[0] / (float)((size_t)NT * CODE_DIM);
  __syncthreads();

  float e = 0.f;
  for (int i = t; i < NB_CODE; i += 256) {
    float p = (float)counts[i] / (float)NT;
    e += p * logf(p + 1e-7f);
  }
  red[t] = e;
  __syncthreads();
  for (int k = 128; k > 0; k >>= 1) { if (t < k) red[t] += red[t + k]; __syncthreads(); }

  if (t == 0) {
    out[(size_t)NT * CODE_DIM]     = commit;       // commit_loss
    out[(size_t)NT * CODE_DIM + 1] = expf(-red[0]); // perplexity
  }
}

// ---------------------------------------------------------------------------
extern "C" void kernel_launch(void* const* d_in, const int* in_sizes, int n_in,
                              void* d_out, int out_size, void* d_ws, size_t ws_size,
                              hipStream_t stream) {
  (void)in_sizes; (void)n_in; (void)out_size; (void)ws_size;
  const float* x  = (const float*)d_in[0];   // [32, 8192, 64] fp32
  const float* cb = (const float*)d_in[1];   // [1024, 64] fp32
  float* out = (float*)d_out;                // [NT*64] x_out, then 2 scalars

  float*    knorm   = (float*)d_ws;                    // 1024 f32
  unsigned* counts  = (unsigned*)d_ws + NB_CODE;       // 1024 u32
  float*    partial = (float*)d_ws + 2 * NB_CODE;      // 2048 f32

  hipMemsetAsync(counts, 0, NB_CODE * sizeof(unsigned), stream);
  knorm_kernel<<<(NB_CODE + 255) / 256, 256, 0, stream>>>(cb, knorm);
  vq_argmin_kernel<<<NBLOCKS, 256, 0, stream>>>(x, cb, knorm, out, counts, partial);
  finalize_kernel<<<1, 256, 0, stream>>>(partial, counts, out);
}